// GATNodeEncoder_2516850835601
// MI455X (gfx1250) — compile-verified
//
#include <hip/hip_runtime.h>
#include <math.h>

typedef float v2f __attribute__((ext_vector_type(2)));
typedef float v8f __attribute__((ext_vector_type(8)));

#define NEG_SLOPE 0.2f
#define EPS_BN 1e-5f

// Sign-aware float atomic max (monotone bit trick; emax initialized to -inf).
__device__ __forceinline__ void atomicMaxFloat(float* addr, float val) {
  if (val >= 0.0f) {
    atomicMax((int*)addr, __float_as_int(val));
  } else {
    atomicMin((unsigned int*)addr, __float_as_uint(val));
  }
}

// ---------------------------------------------------------------------------
// GEMM: C[M,Nc] = A[M,K] @ B[K,Nc], fp32 exact, V_WMMA_F32_16X16X4_F32.
// One wave per 16x64 output tile: 4 accumulators share one A fragment per
// K-step (4x register reuse of A, 4 independent WMMAs per step -> no D->A/B
// hazard stalls). Layouts per CDNA5 ISA 7.12.2:
//   A 16x4 : lanes 0-15 hold M=lane, VGPR0/1 = K{0,1}; lanes 16-31 = K{2,3}
//   B 4x16 : VGPR v, half hf -> row (2*hf + v), col = lane&15
//   C 16x16: VGPR r -> row r (lanes 0-15) / row r+8 (lanes 16-31)
// ---------------------------------------------------------------------------
__global__ void gat_gemm_f32_wmma(const float* __restrict__ A,
                                  const float* __restrict__ B,
                                  float* __restrict__ C,
                                  int M, int K, int Nc) {
  int gid  = blockIdx.x * blockDim.x + threadIdx.x;
  int wave = gid >> 5;
  int lane = threadIdx.x & 31;
  int ntn  = Nc >> 6;                       // 64-wide column super-tiles
  int ntm  = M >> 4;
  if (wave >= ntm * ntn) return;            // wave-uniform exit (EXEC stays full)
  int tm   = wave / ntn;
  int tn   = wave - tm * ntn;
  int row0 = tm << 4;
  int col0 = tn << 6;
  int half = lane >> 4;                     // 0 or 1
  int l    = lane & 15;

  const float* arow  = A + (size_t)(row0 + l) * K + (half << 1);
  const float* bbase = B + (size_t)(half << 1) * Nc + col0 + l;

  v8f acc0 = {0.f, 0.f, 0.f, 0.f, 0.f, 0.f, 0.f, 0.f};
  v8f acc1 = acc0, acc2 = acc0, acc3 = acc0;

  for (int k = 0; k < K; k += 4) {
    v2f a;
    a.x = arow[k];
    a.y = arow[k + 1];
    const float* bp = bbase + (size_t)k * Nc;
    v2f b0, b1, b2, b3;
    b0.x = bp[0];       b0.y = bp[Nc];
    b1.x = bp[16];      b1.y = bp[Nc + 16];
    b2.x = bp[32];      b2.y = bp[Nc + 32];
    b3.x = bp[48];      b3.y = bp[Nc + 48];
    acc0 = __builtin_amdgcn_wmma_f32_16x16x4_f32(false, a, false, b0, (short)0, acc0, false, false);
    acc1 = __builtin_amdgcn_wmma_f32_16x16x4_f32(false, a, false, b1, (short)0, acc1, false, false);
    acc2 = __builtin_amdgcn_wmma_f32_16x16x4_f32(false, a, false, b2, (short)0, acc2, false, false);
    acc3 = __builtin_amdgcn_wmma_f32_16x16x4_f32(false, a, false, b3, (short)0, acc3, false, false);
  }

  float* cp = C + (size_t)(row0 + (half << 3)) * Nc + col0 + l;
#pragma unroll
  for (int r = 0; r < 8; ++r) {
    float* rowp = cp + (size_t)r * Nc;
    rowp[0]  = acc0[r];
    rowp[16] = acc1[r];
    rowp[32] = acc2[r];
    rowp[48] = acc3[r];
  }
}

// ---------------------------------------------------------------------------
// Per-node attention scores: s[n,h] = <h[n,h,:], a_src[h,:]>, d likewise.
// One wave per (node, head); C = 128 -> float4 per lane + wave reduction.
// ---------------------------------------------------------------------------
__global__ void gat_scores(const float* __restrict__ h,
                           const float* __restrict__ asrc,
                           const float* __restrict__ adst,
                           float* __restrict__ s, float* __restrict__ d,
                           int n_nodes, int H, int C) {
  int gid  = blockIdx.x * blockDim.x + threadIdx.x;
  int wave = gid >> 5;
  int lane = threadIdx.x & 31;
  if (wave >= n_nodes * H) return;
  int n  = wave / H;
  int hh = wave - n * H;

  const float4* hp = (const float4*)(h + (size_t)n * H * C + (size_t)hh * C);
  const float4* ap = (const float4*)(asrc + (size_t)hh * C);
  const float4* bp = (const float4*)(adst + (size_t)hh * C);
  float4 hv = hp[lane], av = ap[lane], bv = bp[lane];
  float sv = hv.x * av.x + hv.y * av.y + hv.z * av.z + hv.w * av.w;
  float dv = hv.x * bv.x + hv.y * bv.y + hv.z * bv.z + hv.w * bv.w;
#pragma unroll
  for (int off = 16; off > 0; off >>= 1) {
    sv += __shfl_down(sv, off, 32);
    dv += __shfl_down(dv, off, 32);
  }
  if (lane == 0) { s[wave] = sv; d[wave] = dv; }
}

// Zero/init per-layer scratch: y accumulator, emax(-inf), denom, BN stats.
__global__ void gat_init(float* __restrict__ y, float* __restrict__ emax,
                         float* __restrict__ denom, float* __restrict__ bnsum,
                         float* __restrict__ bnsq, int yN, int nh) {
  int i = blockIdx.x * blockDim.x + threadIdx.x;
  if (i < yN) y[i] = 0.0f;
  if (i < nh) {
    emax[i]  = __int_as_float(0xFF800000);  // -inf
    denom[i] = 0.0f;
  }
  if (i < 128) { bnsum[i] = 0.0f; bnsq[i] = 0.0f; }
}

// Pass 1: segment max of leaky-relu edge scores over dst.
__global__ void gat_edge_max(const int* __restrict__ esrc, const int* __restrict__ edst,
                             const float* __restrict__ s, const float* __restrict__ d,
                             float* __restrict__ emax, int E, int n_nodes, int H) {
  int e = blockIdx.x * blockDim.x + threadIdx.x;
  int T = E + n_nodes;
  if (e >= T) return;
  int u = (e < E) ? esrc[e] : (e - E);      // self-loops appended
  int v = (e < E) ? edst[e] : (e - E);
  for (int hh = 0; hh < H; ++hh) {
    float val = s[u * H + hh] + d[v * H + hh];
    val = (val > 0.0f) ? val : NEG_SLOPE * val;
    atomicMaxFloat(&emax[v * H + hh], val);
  }
}

// Pass 2: segment sum of exp(e - emax[dst]).
__global__ void gat_edge_denom(const int* __restrict__ esrc, const int* __restrict__ edst,
                               const float* __restrict__ s, const float* __restrict__ d,
                               const float* __restrict__ emax, float* __restrict__ denom,
                               int E, int n_nodes, int H) {
  int e = blockIdx.x * blockDim.x + threadIdx.x;
  int T = E + n_nodes;
  if (e >= T) return;
  int u = (e < E) ? esrc[e] : (e - E);
  int v = (e < E) ? edst[e] : (e - E);
  for (int hh = 0; hh < H; ++hh) {
    float val = s[u * H + hh] + d[v * H + hh];
    val = (val > 0.0f) ? val : NEG_SLOPE * val;
    float ex = expf(val - emax[v * H + hh]);
    atomicAdd(&denom[v * H + hh], ex);
  }
}

// Pass 3: y[dst,c] += sum_h (alpha_h/H) * h[src,h,c].  One wave per edge,
// looping heads internally (src/dst/score gathers loaded once per edge).
// 32 lanes x float4 cover C=128. Head-mean folded in via invH.
__global__ void gat_edge_scatter(const int* __restrict__ esrc, const int* __restrict__ edst,
                                 const float* __restrict__ s, const float* __restrict__ d,
                                 const float* __restrict__ emax, const float* __restrict__ denom,
                                 const float* __restrict__ h, float* __restrict__ y,
                                 int E, int n_nodes, int H, int C, float invH) {
  int gid  = blockIdx.x * blockDim.x + threadIdx.x;
  int wave = gid >> 5;
  int lane = threadIdx.x & 31;
  int T = E + n_nodes;
  if (wave >= T) return;
  int e = wave;
  int u = (e < E) ? esrc[e] : (e - E);
  int v = (e < E) ? edst[e] : (e - E);

  float* yp = y + (size_t)v * C + lane * 4;
  const float4* hp = (const float4*)(h + (size_t)u * H * C);

  for (int hh = 0; hh < H; ++hh) {
    float val = s[u * H + hh] + d[v * H + hh];
    val = (val > 0.0f) ? val : NEG_SLOPE * val;
    float alpha = expf(val - emax[v * H + hh]) / (denom[v * H + hh] + 1e-16f) * invH;

    float4 hv = hp[(hh * C >> 2) + lane];
    atomicAdd(yp + 0, hv.x * alpha);
    atomicAdd(yp + 1, hv.y * alpha);
    atomicAdd(yp + 2, hv.z * alpha);
    atomicAdd(yp + 3, hv.w * alpha);
  }
}

// BN stats: per-channel sum and sum-of-squares (block = 128 channels).
__global__ void gat_bn_stats(const float* __restrict__ y, float* __restrict__ bnsum,
                             float* __restrict__ bnsq, int n_nodes, int C) {
  int c = threadIdx.x;                       // blockDim.x == C
  float s0 = 0.0f, s1 = 0.0f;
  for (int r = blockIdx.x; r < n_nodes; r += gridDim.x) {
    float v = y[(size_t)r * C + c];
    s0 += v;
    s1 += v * v;
  }
  atomicAdd(&bnsum[c], s0);
  atomicAdd(&bnsq[c], s1);
}

// BN apply (+optional ReLU). Bias b cancels under BN (constant shift), omitted.
__global__ void gat_bn_apply(float* __restrict__ y, const float* __restrict__ bnsum,
                             const float* __restrict__ bnsq, const float* __restrict__ gamma,
                             const float* __restrict__ beta, int n_nodes, int C, int do_relu) {
  long long idx = (long long)blockIdx.x * blockDim.x + threadIdx.x;
  if (idx >= (long long)n_nodes * C) return;
  int c = (int)(idx % C);
  float inv_n = 1.0f / (float)n_nodes;
  float mu  = bnsum[c] * inv_n;
  float var = bnsq[c] * inv_n - mu * mu;
  float v = (y[idx] - mu) * rsqrtf(var + EPS_BN) * gamma[c] + beta[c];
  if (do_relu) v = fmaxf(v, 0.0f);
  y[idx] = v;
}

// ---------------------------------------------------------------------------
extern "C" void kernel_launch(void* const* d_in, const int* in_sizes, int n_in,
                              void* d_out, int out_size, void* d_ws, size_t ws_size,
                              hipStream_t stream) {
  const float* x    = (const float*)d_in[0];       // [N, 256]
  const int*   eidx = (const int*)d_in[1];         // [2, E]
  int n_nodes = in_sizes[0] / 256;
  int E       = in_sizes[1] / 2;
  const int* esrc = eidx;
  const int* edst = eidx + E;

  // Workspace layout (floats): h[N*512] | y[N*128] | s[N*4] | d[N*4] |
  //                            emax[N*4] | denom[N*4] | bnsum[128] | bnsq[128]
  float* ws    = (float*)d_ws;
  float* h     = ws;
  float* ybuf  = h     + (size_t)n_nodes * 512;
  float* sarr  = ybuf  + (size_t)n_nodes * 128;
  float* darr  = sarr  + (size_t)n_nodes * 4;
  float* emax  = darr  + (size_t)n_nodes * 4;
  float* denom = emax  + (size_t)n_nodes * 4;
  float* bnsum = denom + (size_t)n_nodes * 4;
  float* bnsq  = bnsum + 128;

  struct LayerP { const float *W, *as, *ad, *g, *be; int K, H, C; };
  LayerP L[3] = {
      {(const float*)d_in[2],  (const float*)d_in[3],  (const float*)d_in[4],
       (const float*)d_in[6],  (const float*)d_in[7],  256, 4, 128},
      {(const float*)d_in[8],  (const float*)d_in[9],  (const float*)d_in[10],
       (const float*)d_in[12], (const float*)d_in[13], 128, 4, 128},
      {(const float*)d_in[14], (const float*)d_in[15], (const float*)d_in[16],
       (const float*)d_in[18], (const float*)d_in[19], 128, 1, 128},
  };

  const float* in = x;
  int T = E + n_nodes;  // edges incl. self-loops

  for (int i = 0; i < 3; ++i) {
    int H = L[i].H, C = L[i].C, K = L[i].K;
    int HC = H * C;
    float* ytgt = (i == 2) ? (float*)d_out : ybuf;

    // 1) h = in @ W  (fp32 WMMA, 16x64 wave tiles)
    int tiles = (n_nodes / 16) * (HC / 64);
    gat_gemm_f32_wmma<<<(tiles + 3) / 4, 128, 0, stream>>>(in, L[i].W, h, n_nodes, K, HC);

    // 2) per-node scores
    int swaves = n_nodes * H;
    gat_scores<<<(swaves * 32 + 255) / 256, 256, 0, stream>>>(h, L[i].as, L[i].ad,
                                                              sarr, darr, n_nodes, H, C);

    // 3) init accumulators for this layer
    int initN = n_nodes * C;
    gat_init<<<(initN + 255) / 256, 256, 0, stream>>>(ytgt, emax, denom, bnsum, bnsq,
                                                      initN, n_nodes * H);

    // 4) segment softmax + scatter
    gat_edge_max<<<(T + 255) / 256, 256, 0, stream>>>(esrc, edst, sarr, darr, emax,
                                                      E, n_nodes, H);
    gat_edge_denom<<<(T + 255) / 256, 256, 0, stream>>>(esrc, edst, sarr, darr, emax,
                                                        denom, E, n_nodes, H);
    long long sthreads = (long long)T * 32;   // one wave per edge
    gat_edge_scatter<<<(unsigned)((sthreads + 255) / 256), 256, 0, stream>>>(
        esrc, edst, sarr, darr, emax, denom, h, ytgt, E, n_nodes, H, C, 1.0f / (float)H);

    // 5) BatchNorm (+ReLU except last layer)
    gat_bn_stats<<<512, 128, 0, stream>>>(ytgt, bnsum, bnsq, n_nodes, C);
    long long elems = (long long)n_nodes * C;
    gat_bn_apply<<<(unsigned)((elems + 255) / 256), 256, 0, stream>>>(
        ytgt, bnsum, bnsq, L[i].g, L[i].be, n_nodes, C, (i < 2) ? 1 : 0);

    in = ytgt;
  }
}